// DSSIMLoss_13950053777961
// MI455X (gfx1250) — compile-verified
//
#include <hip/hip_runtime.h>

typedef float v8f __attribute__((ext_vector_type(8)));
typedef float v2f __attribute__((ext_vector_type(2)));

#define IMG_H 512
#define IMG_W 512
#define N_IMG 96            // 32 batch * 3 channels
#define OH 507              // 512 - 6 + 1
#define OW 507
#define TH 32               // output tile height per block
#define TW 48               // output tile width per block
#define IN_H (TH + 5)       // 37
#define IN_W (TW + 5)       // 53
#define PAD_W (IN_W + 1)    // 54 (pad to dodge LDS bank conflicts)
#define NTHR 192            // 6 wave32
#define TILES_H 16          // ceil(507/32)
#define TILES_W 11          // ceil(507/48)
#define NSLOTS 512

// 1-D gaussian, w=6, sigma=1.5, normalized (matches reference _gaussian_window)
#define G0 0.03802583f
#define G1 0.11551229f
#define G2 0.22498722f
#define G3 0.28097498f
#define G4 G2
#define G5 G1

__device__ __forceinline__ float gsel(int t) {
    float r = 0.0f;
    r = (t == 0) ? G0 : r;
    r = (t == 1) ? G1 : r;
    r = (t == 2) ? G2 : r;
    r = (t == 3) ? G3 : r;
    r = (t == 4) ? G4 : r;
    r = (t == 5) ? G5 : r;
    return r;
}

// Vertical 6-tap gaussian over a 16x16 tile expressed as 6 chained
// V_WMMA_F32_16X16X4_F32 ops:  D[m][n] = sum_j sum_k G[m][4j+k] * R[h0+4j+k][n]
// with G the 16x(21..24) band matrix of gaussian weights (zeros past tap 5).
__device__ __forceinline__ v8f conv_col(const float* __restrict__ r,
                                        const v2f* aW, int h0s, int khalf, int col) {
    v8f c = {};
#pragma unroll
    for (int j = 0; j < 6; ++j) {
        int r0 = h0s + 4 * j + khalf;
        int r1 = r0 + 1;
        if (r0 > IN_H - 1) r0 = IN_H - 1;   // rows past the band pair with 0-weights
        if (r1 > IN_H - 1) r1 = IN_H - 1;
        v2f b;
        b.x = r[r0 * TW + col];
        b.y = r[r1 * TW + col];
        c = __builtin_amdgcn_wmma_f32_16x16x4_f32(false, aW[j], false, b,
                                                  (short)0, c, false, false);
    }
    return c;
}

__global__ void k_init(int* wsi, double* slots) {
    int t = threadIdx.x;
    if (t < 2) wsi[t] = 0;
    if (t < NSLOTS) slots[t] = 0.0;
}

__global__ void k_flags(const float4* __restrict__ x, int n4, int* wsi) {
    int stride = gridDim.x * blockDim.x;
    bool hi = false, lo = false;
    for (int i = blockIdx.x * blockDim.x + threadIdx.x; i < n4; i += stride) {
        float4 v = x[i];
        hi |= (v.x > 128.0f) | (v.y > 128.0f) | (v.z > 128.0f) | (v.w > 128.0f);
        lo |= (v.x < -0.5f) | (v.y < -0.5f) | (v.z < -0.5f) | (v.w < -0.5f);
    }
    if (hi) atomicOr(wsi + 0, 1);
    if (lo) atomicOr(wsi + 1, 1);
}

__global__ void __launch_bounds__(NTHR) k_ssim(const float* __restrict__ x,
                                               const float* __restrict__ y,
                                               const int* __restrict__ flags,
                                               double* __restrict__ slots) {
    __shared__ float sx[IN_H][PAD_W];
    __shared__ float sy[IN_H][PAD_W];
    __shared__ float rA [IN_H * TW];   // gauss_row(x)
    __shared__ float rB [IN_H * TW];   // gauss_row(y)
    __shared__ float rXX[IN_H * TW];   // gauss_row(x*x)
    __shared__ float rYY[IN_H * TW];   // gauss_row(y*y)
    __shared__ float rXY[IN_H * TW];   // gauss_row(x*y)

    const int tid = threadIdx.x;
    const int bid = blockIdx.x;
    const int img = bid / (TILES_H * TILES_W);
    const int rem = bid % (TILES_H * TILES_W);
    const int gh0 = (rem / TILES_W) * TH;
    const int gw0 = (rem % TILES_W) * TW;

    const float* xb = x + (size_t)img * (IMG_H * IMG_W);
    const float* yb = y + (size_t)img * (IMG_H * IMG_W);

    // ---- Stage 1: coalesced raw-tile load (edge-clamped; excess is masked later)
    for (int p = tid; p < IN_H * IN_W; p += NTHR) {
        int rr = p / IN_W, cc = p % IN_W;
        int gh = gh0 + rr; gh = gh > IMG_H - 1 ? IMG_H - 1 : gh;
        int gw = gw0 + cc; gw = gw > IMG_W - 1 ? IMG_W - 1 : gw;
        size_t o = (size_t)gh * IMG_W + gw;
        sx[rr][cc] = xb[o];
        sy[rr][cc] = yb[o];
    }
    __syncthreads();

    // ---- Stage 2: horizontal 6-tap gaussian of 5 quantities
    for (int p = tid; p < IN_H * TW; p += NTHR) {
        int h = p / TW, w = p % TW;
        float ax = 0.f, ay = 0.f, axx = 0.f, ayy = 0.f, axy = 0.f;
#pragma unroll
        for (int k = 0; k < 6; ++k) {
            const float gk = gsel(k);            // const-folds per unrolled k
            float xv = sx[h][w + k];
            float yv = sy[h][w + k];
            ax  += gk * xv;
            ay  += gk * yv;
            axx += gk * xv * xv;
            ayy += gk * yv * yv;
            axy += gk * xv * yv;
        }
        rA[p] = ax; rB[p] = ay; rXX[p] = axx; rYY[p] = ayy; rXY[p] = axy;
    }
    __syncthreads();

    // ---- Stage 3: vertical tap via WMMA, one 16x16 subtile per wave
    const int lane  = tid & 31;
    const int wave  = tid >> 5;              // 0..5
    const int w0s   = (wave % 3) * 16;
    const int h0s   = (wave / 3) * 16;
    const int m     = lane & 15;             // A-row / B-col / D-col index
    const int khalf = (lane >> 4) * 2;       // K half owned by this lane
    const int col   = w0s + m;

    v2f aW[6];                               // shared A for all 5 quantities
#pragma unroll
    for (int j = 0; j < 6; ++j) {
        aW[j].x = gsel(4 * j + khalf - m);
        aW[j].y = gsel(4 * j + khalf + 1 - m);
    }

    v8f cMu1 = conv_col(rA,  aW, h0s, khalf, col);
    v8f cMu2 = conv_col(rB,  aW, h0s, khalf, col);
    v8f cXX  = conv_col(rXX, aW, h0s, khalf, col);
    v8f cYY  = conv_col(rYY, aW, h0s, khalf, col);
    v8f cXY  = conv_col(rXY, aW, h0s, khalf, col);

    // dynamic range constants (uniform scalar loads)
    float maxv = flags[0] ? 255.0f : 1.0f;
    float minv = flags[1] ? -1.0f : 0.0f;
    float Lr = maxv - minv;
    float C1 = (0.01f * Lr) * (0.01f * Lr);
    float C2 = (0.03f * Lr) * (0.03f * Lr);

    float lsum = 0.0f;
#pragma unroll
    for (int i = 0; i < 8; ++i) {
        int M  = i + ((lane >> 4) << 3);     // C/D layout: VGPR i -> M = i (+8 hi half)
        int gh = gh0 + h0s + M;
        int gw = gw0 + col;
        float mu1 = cMu1[i], mu2 = cMu2[i];
        float mu1s = mu1 * mu1, mu2s = mu2 * mu2, mu12 = mu1 * mu2;
        float s1  = cXX[i] - mu1s;
        float s2  = cYY[i] - mu2s;
        float s12 = cXY[i] - mu12;
        float v1  = 2.0f * s12 + C2;
        float v2  = s1 + s2 + C2;
        float num = (2.0f * mu12 + C1) * v1;
        float den = (mu1s + mu2s + C1) * v2;
        float val = num / den;
        if (gh < OH && gw < OW) lsum += val;
    }

    // wave32 shuffle reduce, one f64 atomic per wave spread over 512 slots
#pragma unroll
    for (int off = 16; off > 0; off >>= 1)
        lsum += __shfl_down(lsum, off, 32);
    if (lane == 0)
        atomicAdd(&slots[(bid * 6 + wave) & (NSLOTS - 1)], (double)lsum);
}

__global__ void k_final(const double* __restrict__ slots, float* __restrict__ out) {
    __shared__ double red[NSLOTS];
    int t = threadIdx.x;
    red[t] = slots[t];
    __syncthreads();
    for (int s = NSLOTS / 2; s > 0; s >>= 1) {
        if (t < s) red[t] += red[t + s];
        __syncthreads();
    }
    if (t == 0) {
        double mean = red[0] / (double)((double)N_IMG * (double)OH * (double)OW);
        out[0] = (float)((1.0 - mean) * 0.5);
    }
}

extern "C" void kernel_launch(void* const* d_in, const int* in_sizes, int n_in,
                              void* d_out, int out_size, void* d_ws, size_t ws_size,
                              hipStream_t stream) {
    const float* x = (const float*)d_in[0];   // "output" (y_pred)
    const float* y = (const float*)d_in[1];   // "target" (y_true)
    float* out = (float*)d_out;

    int*    wsi   = (int*)d_ws;                       // 2 range flags
    double* slots = (double*)((char*)d_ws + 64);      // 512 f64 accumulators

    k_init<<<1, NSLOTS, 0, stream>>>(wsi, slots);

    int n4 = in_sizes[0] / 4;                         // 6,291,456 float4's
    k_flags<<<4096, 256, 0, stream>>>((const float4*)x, n4, wsi);

    k_ssim<<<N_IMG * TILES_H * TILES_W, NTHR, 0, stream>>>(x, y, wsi, slots);

    k_final<<<1, NSLOTS, 0, stream>>>(slots, out);
}